// TextLoss_12369505812997
// MI455X (gfx1250) — compile-verified
//
#include <hip/hip_runtime.h>
#include <hip/hip_bf16.h>

typedef __attribute__((ext_vector_type(2))) float v2f;
typedef __attribute__((ext_vector_type(8))) float v8f;

#define Bn   8
#define Hn   640
#define Wn   640
#define Npix (Hn * Wn)          // 409600
#define NBINS 1024

// workspace layout (floats)
#define OFF_ACC     0            // 4: S_bce, S_norm, S_angle, mask_cnt
#define OFF_PB_P    4            // 8: per-batch positive count
#define OFF_PB_SP   12           // 8: per-batch sum of dis_pre over pos
#define OFF_PB_SN   20           // 8: per-batch sum of dis_pre over neg
#define OFF_DIS     28           // 1: dis_loss accumulator
#define OFF_POINT   29           // 1: point_loss accumulator
#define OFF_ENERGY  32           // 4*64 energy values
#define OFF_HIST    288          // [4 types][8 batches][NBINS]; types: 0=neg_cnt 1=neg_sum 2=all_cnt 3=all_sum
#define WS_FLOATS   (288 + 4 * 8 * NBINS)

__global__ void zero_ws(float* ws, int n) {
    int i = blockIdx.x * blockDim.x + threadIdx.x;
    if (i < n) ws[i] = 0.0f;
}

// ---------------------------------------------------------------------------
// Main fused pass: 100 blocks/batch x 8 batches, 256 thr, 16 px/thread.
// Accumulates 7 scalars per block (wmma-reduced) + LDS histograms of dis_pre.
// ---------------------------------------------------------------------------
__global__ void __launch_bounds__(256)
main_pass(const float* __restrict__ fy, const unsigned char* __restrict__ tmask,
          const int* __restrict__ trm, const float* __restrict__ df,
          const float* __restrict__ dirf, const float* __restrict__ wm,
          float* __restrict__ ws) {
    __shared__ float h_nc[NBINS], h_ns[NBINS], h_ac[NBINS], h_as[NBINS];
    __shared__ float red[8][256];

    const int tid = threadIdx.x;
    const int cb  = blockIdx.x;       // chunk within batch (4096 px)
    const int b   = blockIdx.y;

    for (int i = tid; i < NBINS; i += 256) { h_nc[i] = 0.f; h_ns[i] = 0.f; h_ac[i] = 0.f; h_as[i] = 0.f; }
    __syncthreads();

    const float* p0  = fy + (size_t)(b * 4 + 0) * Npix;
    const float* p1  = fy + (size_t)(b * 4 + 1) * Npix;
    const float* p2  = fy + (size_t)(b * 4 + 2) * Npix;
    const float* p3  = fy + (size_t)(b * 4 + 3) * Npix;
    const float* dfb = df + (size_t)b * Npix;
    const float* gxb = dirf + (size_t)(b * 2 + 0) * Npix;
    const float* gyb = dirf + (size_t)(b * 2 + 1) * Npix;
    const float* wmb = wm + (size_t)b * Npix;
    const unsigned char* tmb = tmask + (size_t)b * Npix;
    const int* trb = trm + (size_t)b * Npix;

    float aBce = 0.f, aNorm = 0.f, aAng = 0.f, aCnt = 0.f, aP = 0.f, aSp = 0.f, aSn = 0.f;

    for (int g = 0; g < 4; ++g) {
        const int s = cb * 4096 + g * 1024 + tid * 4;
        float4 vcls = *(const float4*)(p0 + s);
        float4 vf1  = *(const float4*)(p1 + s);
        float4 vfx  = *(const float4*)(p2 + s);
        float4 vfy  = *(const float4*)(p3 + s);
        float4 vdf  = *(const float4*)(dfb + s);
        float4 vgx  = *(const float4*)(gxb + s);
        float4 vgy  = *(const float4*)(gyb + s);
        float4 vw   = *(const float4*)(wmb + s);
        uchar4 vtm  = *(const uchar4*)(tmb + s);
        int4   vtr  = *(const int4*)(trb + s);

        const float* fcls = (const float*)&vcls;
        const float* ff1  = (const float*)&vf1;
        const float* ffx  = (const float*)&vfx;
        const float* ffy  = (const float*)&vfy;
        const float* fdf  = (const float*)&vdf;
        const float* fgx  = (const float*)&vgx;
        const float* fgy  = (const float*)&vgy;
        const float* fw   = (const float*)&vw;
        const unsigned char* ftm = (const unsigned char*)&vtm;
        const int* ftr = (const int*)&vtr;

#pragma unroll
        for (int l = 0; l < 4; ++l) {
            const float tmf = ftm[l] ? 1.f : 0.f;
            const float t   = (ftr[l] > 0) ? 1.f : 0.f;
            // cls BCE
            const float pc = fcls[l];
            aBce += -(t * __logf(pc) + (1.f - t) * __logf(1.f - pc)) * tmf;
            // distance term + histogram
            const float dfl = fdf[l];
            const float d1  = ff1[l] - dfl;
            const float dis_pre = d1 * d1 * tmf;
            const bool pos = (dfl >= 0.001f);
            int bin = (int)(dis_pre * (float)NBINS);
            bin = bin > (NBINS - 1) ? (NBINS - 1) : bin;
            atomicAdd(&h_ac[bin], 1.f);
            atomicAdd(&h_as[bin], dis_pre);
            if (pos) { aP += 1.f; aSp += dis_pre; }
            else     { aSn += dis_pre; atomicAdd(&h_nc[bin], 1.f); atomicAdd(&h_ns[bin], dis_pre); }
            // flux norm
            const float gx = fgx[l], gy = fgy[l];
            const float gs = 0.999999f / (sqrtf(gx * gx + gy * gy) + 0.001f);
            const float gxn = gx * gs, gyn = gy * gs;
            const float fx = ffx[l], fyv = ffy[l];
            const float ex = fx - gxn, ey = fyv - gyn;
            aNorm += fw[l] * 0.5f * (ex * ex + ey * ey) * tmf;
            // angle
            const float ps2 = 0.999999f / (sqrtf(fx * fx + fyv * fyv) + 0.001f);
            const float pxn = fx * ps2, pyn = fyv * ps2;
            const float num = pxn * gxn + pyn * gyn;
            const float den = fmaxf(sqrtf(pxn * pxn + pyn * pyn) * sqrtf(gxn * gxn + gyn * gyn), 1e-8f);
            const float m = ((ftm[l] != 0) && (ftr[l] > 0)) ? 1.f : 0.f;
            aAng += (1.f - num / den) * m;
            aCnt += m;
        }
    }

    // stage partials quantity-major for the wmma reduction
    red[0][tid] = aBce; red[1][tid] = aNorm; red[2][tid] = aAng; red[3][tid] = aCnt;
    red[4][tid] = aP;   red[5][tid] = aSp;   red[6][tid] = aSn;  red[7][tid] = 0.f;
    __syncthreads();

    // flush LDS histograms to global per-batch histograms
    float* g_nc = ws + OFF_HIST + (size_t)(0 * 8 + b) * NBINS;
    float* g_ns = ws + OFF_HIST + (size_t)(1 * 8 + b) * NBINS;
    float* g_ac = ws + OFF_HIST + (size_t)(2 * 8 + b) * NBINS;
    float* g_as = ws + OFF_HIST + (size_t)(3 * 8 + b) * NBINS;
    for (int i = tid; i < NBINS; i += 256) {
        if (h_nc[i] != 0.f) atomicAdd(&g_nc[i], h_nc[i]);
        if (h_ns[i] != 0.f) atomicAdd(&g_ns[i], h_ns[i]);
        if (h_ac[i] != 0.f) atomicAdd(&g_ac[i], h_ac[i]);
        if (h_as[i] != 0.f) atomicAdd(&g_as[i], h_as[i]);
    }

    // wave 0 reduces all 8 quantities via chained V_WMMA_F32_16X16X4_F32:
    // D = A(16x4 data) * B(4x16 ones) + C  => row-sums accumulate over chunks.
    if (tid < 32) {
        v2f bones; bones.x = 1.0f; bones.y = 1.0f;
        float res[8];
#pragma unroll
        for (int q = 0; q < 8; ++q) {
            v8f c = {};
#pragma unroll
            for (int ch = 0; ch < 4; ++ch) {
                v2f a;
                a.x = red[q][ch * 64 + tid];
                a.y = red[q][ch * 64 + 32 + tid];
                c = __builtin_amdgcn_wmma_f32_16x16x4_f32(
                        false, a, false, bones, (short)0, c, false, false);
            }
            // lanes 0-15 hold rows 0..7 sums across the 8 D VGPRs, lanes 16-31 rows 8..15
            float ssum = c[0] + c[1] + c[2] + c[3] + c[4] + c[5] + c[6] + c[7];
            ssum += __shfl_xor(ssum, 16, 32);
            res[q] = ssum;
        }
        if (tid == 0) {
            atomicAdd(&ws[OFF_ACC + 0], res[0]);
            atomicAdd(&ws[OFF_ACC + 1], res[1]);
            atomicAdd(&ws[OFF_ACC + 2], res[2]);
            atomicAdd(&ws[OFF_ACC + 3], res[3]);
            atomicAdd(&ws[OFF_PB_P + b], res[4]);
            atomicAdd(&ws[OFF_PB_SP + b], res[5]);
            atomicAdd(&ws[OFF_PB_SN + b], res[6]);
        }
    }
}

// ---------------------------------------------------------------------------
// Per-batch dis-loss finalize: histogram top-k walk replaces the sort.
// ---------------------------------------------------------------------------
__global__ void dis_finalize(float* __restrict__ ws) {
    if (threadIdx.x != 0) return;
    const int b = blockIdx.x;
    const float* g_nc = ws + OFF_HIST + (size_t)(0 * 8 + b) * NBINS;
    const float* g_ns = ws + OFF_HIST + (size_t)(1 * 8 + b) * NBINS;
    const float* g_ac = ws + OFF_HIST + (size_t)(2 * 8 + b) * NBINS;
    const float* g_as = ws + OFF_HIST + (size_t)(3 * 8 + b) * NBINS;

    const float P  = ws[OFF_PB_P + b];
    const float Nn = (float)Npix - P;
    const float posi = ws[OFF_PB_SP + b] / fmaxf(P, 1.f);
    const float mean_neg = ws[OFF_PB_SN + b] / fmaxf(Nn, 1.f);
    const float k  = 3.f * P;
    const float kk = fminf(k, Nn);

    float cum = 0.f, tsum = 0.f;
    for (int bin = NBINS - 1; bin >= 0; --bin) {
        const float c = g_nc[bin];
        if (cum + c <= kk) { tsum += g_ns[bin]; cum += c; }
        else {
            const float need = kk - cum;
            if (need > 0.f && c > 0.f) tsum += g_ns[bin] * (need / c);
            break;
        }
    }
    const float topk_mean = tsum / fmaxf(k, 1.f);
    const float nega = (Nn < k) ? mean_neg : topk_mean;

    float cum2 = 0.f, t100 = 0.f;
    for (int bin = NBINS - 1; bin >= 0; --bin) {
        const float c = g_ac[bin];
        if (cum2 + c <= 100.f) { t100 += g_as[bin]; cum2 += c; }
        else {
            const float need = 100.f - cum2;
            if (need > 0.f && c > 0.f) t100 += g_as[bin] * (need / c);
            break;
        }
    }
    const float top100 = t100 * 0.01f;
    const float contrib = (P > 0.5f) ? (posi + nega) : top100;
    atomicAdd(&ws[OFF_DIS], contrib * (1.f / (float)Bn));
}

// ---------------------------------------------------------------------------
// Point loss: grid (64 polys, 3 iters), 32 threads; lane s computes shift s.
// ---------------------------------------------------------------------------
__global__ void point_loss(const float* __restrict__ py, const int* __restrict__ inds,
                           const float* __restrict__ gtp, float* __restrict__ ws) {
    const int j  = blockIdx.x;
    const int it = blockIdx.y + 1;
    const int s  = threadIdx.x;
    const int b0 = inds[j];
    const int a0 = inds[64 + j];
    const float* gt = gtp + ((size_t)(b0 * 8 + a0) * 20) * 2;
    const float* pp = py + ((size_t)(it * 64 + j) * 20) * 2;

    float d = 3.4e38f;
    if (s < 20) {
        float acc = 0.f;
#pragma unroll
        for (int p = 0; p < 20; ++p) {
            const int gi = (p + s) % 20;
            const float dx = pp[2 * p]     - gt[2 * gi];
            const float dy = pp[2 * p + 1] - gt[2 * gi + 1];
            acc += sqrtf(dx * dx + dy * dy);
        }
        d = acc;
    }
#pragma unroll
    for (int off = 16; off >= 1; off >>= 1) d = fminf(d, __shfl_xor(d, off, 32));
    if (s == 0) atomicAdd(&ws[OFF_POINT], d * (1.f / (64.f * 3.f)));
}

// ---------------------------------------------------------------------------
// Energy: bilinear sample of distance_field at all (iter, poly) positions.
// ---------------------------------------------------------------------------
__global__ void energy_pass(const float* __restrict__ py, const int* __restrict__ inds,
                            const float* __restrict__ df, float* __restrict__ ws) {
    const int tid = threadIdx.x;     // 0..255
    const int it = tid >> 6;
    const int j  = tid & 63;
    const int b0 = inds[j];
    const float* field = df + (size_t)b0 * Npix;
    const float* pp = py + ((size_t)(it * 64 + j) * 20) * 2;

    float sum = 0.f;
    for (int p = 0; p < 20; ++p) {
        const float px = pp[2 * p], pv = pp[2 * p + 1];
        // with h=w=H=W: ix = px - 0.5, iy = py - 0.5
        const float ix = px - 0.5f, iy = pv - 0.5f;
        const float x0 = floorf(ix), y0 = floorf(iy);
        const float x1 = x0 + 1.f, y1 = y0 + 1.f;
        const float fx1 = x1 - ix, fx0 = ix - x0;
        const float fy1 = y1 - iy, fy0 = iy - y0;
        auto tap = [&](float yy, float xx, float wgt) -> float {
            const bool valid = (xx >= 0.f) && (xx < (float)Wn) && (yy >= 0.f) && (yy < (float)Hn);
            const int xi = (int)fminf(fmaxf(xx, 0.f), (float)(Wn - 1));
            const int yi = (int)fminf(fmaxf(yy, 0.f), (float)(Hn - 1));
            const float v = field[yi * Wn + xi];
            return (valid ? v : 0.f) * wgt;
        };
        sum += tap(y0, x0, fx1 * fy1) + tap(y0, x1, fx0 * fy1)
             + tap(y1, x0, fx1 * fy0) + tap(y1, x1, fx0 * fy0);
    }
    ws[OFF_ENERGY + it * 64 + j] = sum;
}

// ---------------------------------------------------------------------------
// Final combine into the scalar output.
// ---------------------------------------------------------------------------
__global__ void final_combine(const float* __restrict__ ws, float* __restrict__ out) {
    const float cls   = ws[OFF_ACC + 0] / ((float)Bn * (float)Npix);
    const float nrm   = ws[OFF_ACC + 1] / ((float)Bn * (float)Hn);
    const float ang   = ws[OFF_ACC + 2] / fmaxf(ws[OFF_ACC + 3], 1.f);
    const float dis   = ws[OFF_DIS];
    const float point = ws[OFF_POINT];

    float reg = 0.f, en = 0.f;
    for (int i = 1; i < 4; ++i) {
        for (int j = 0; j < 64; ++j) {
            const float e1 = ws[OFF_ENERGY + i * 64 + j];
            const float e0 = ws[OFF_ENERGY + (i - 1) * 64 + j];
            reg += fmaxf(e1 - e0, 0.f) * (1.f / 64.f);
            en  += (e1 <= 0.01f ? 0.f : e1) * (1.f / 64.f);
        }
    }
    const float energy_loss = (en + reg) / 3.f;
    out[0] = 1.0f * cls + 3.0f * dis + 0.5f * (nrm + ang) + 0.05f * (point + energy_loss);
}

extern "C" void kernel_launch(void* const* d_in, const int* in_sizes, int n_in,
                              void* d_out, int out_size, void* d_ws, size_t ws_size,
                              hipStream_t stream) {
    const float* fy   = (const float*)d_in[0];
    const float* py   = (const float*)d_in[1];
    const int*   inds = (const int*)d_in[2];
    const unsigned char* tmask = (const unsigned char*)d_in[3];
    const int*   trm  = (const int*)d_in[4];
    const float* df   = (const float*)d_in[5];
    const float* dirf = (const float*)d_in[6];
    const float* wm   = (const float*)d_in[7];
    const float* gtp  = (const float*)d_in[8];
    float* ws  = (float*)d_ws;
    float* out = (float*)d_out;

    zero_ws<<<(WS_FLOATS + 255) / 256, 256, 0, stream>>>(ws, WS_FLOATS);
    main_pass<<<dim3(Npix / 4096, Bn), 256, 0, stream>>>(fy, tmask, trm, df, dirf, wm, ws);
    dis_finalize<<<Bn, 32, 0, stream>>>(ws);
    point_loss<<<dim3(64, 3), 32, 0, stream>>>(py, inds, gtp, ws);
    energy_pass<<<1, 256, 0, stream>>>(py, inds, df, ws);
    final_combine<<<1, 1, 0, stream>>>(ws, out);
}